// Res5ROIHeads_70274254897559
// MI455X (gfx1250) — compile-verified
//
#include <hip/hip_runtime.h>
#include <hip/hip_bf16.h>
#include <stdint.h>

// Problem sizes (fixed by the reference).
#define NN 16
#define GG 512
#define PP 4096
#define BB 512      // BATCH_SIZE_PER_IMAGE
#define FG_CAP 128  // BATCH_SIZE_PER_IMAGE * POSITIVE_FRACTION

// Flat offsets (in float elements) of each tuple member inside d_out,
// concatenated in reference return order.
static constexpr size_t OFF_IOU   = 0;
static constexpr size_t OFF_MIDX  = OFF_IOU + (size_t)NN * GG * PP;
static constexpr size_t OFF_CLS   = OFF_MIDX + (size_t)NN * PP;
static constexpr size_t OFF_SINDS = OFF_CLS + (size_t)NN * PP;
static constexpr size_t OFF_SCLS  = OFF_SINDS + (size_t)NN * BB;
static constexpr size_t OFF_SGT   = OFF_SCLS + (size_t)NN * BB;
static constexpr size_t OFF_SPROP = OFF_SGT + (size_t)NN * BB * 4;
static constexpr size_t OFF_VALID = OFF_SPROP + (size_t)NN * BB * 4;

typedef unsigned int u32x4 __attribute__((ext_vector_type(4)));
typedef int          i32x4 __attribute__((ext_vector_type(4)));
typedef int          i32x8 __attribute__((ext_vector_type(8)));
typedef float        f32x4 __attribute__((ext_vector_type(4)));

#if defined(__gfx1250__) && __has_builtin(__builtin_amdgcn_tensor_load_to_lds)
#define HAVE_TDM 1
#else
#define HAVE_TDM 0
#endif

// Issue a 1D TDM copy of n_dwords dwords from global memory to LDS.
// Only wave 0 of the block issues the DMA (TDM ignores EXEC; one issue per wave).
// Probe-verified lowering: tensor_load_to_lds s[g0], s[g1] + s_wait_tensorcnt.
__device__ __forceinline__ void stage_to_lds(void* lds_dst, const void* gsrc,
                                             unsigned n_dwords,
                                             unsigned tid, unsigned nthr) {
#if HAVE_TDM
  (void)nthr;
  if ((tid >> 5) == 0) {
    unsigned long long ga = (unsigned long long)(uintptr_t)gsrc;
    u32x4 g0;
    g0[0] = 1u;                                    // count=1 (valid), user mode, no gather
    g0[1] = (unsigned)(uintptr_t)lds_dst;          // lds_addr: low 32 bits of generic = LDS offset
    g0[2] = (unsigned)(ga & 0xFFFFFFFFu);          // global_addr[31:0]
    g0[3] = (unsigned)((ga >> 32) & 0x01FFFFFFu)   // global_addr[56:32]
          | (2u << 30);                            // type = 2 ("image")
    i32x8 g1;
    g1[0] = (int)(2u << 16);                       // workgroup_mask=0, data_size=2 (4 bytes)
    g1[1] = (int)((n_dwords & 0xFFFFu) << 16);     // tensor_dim0[15:0]
    g1[2] = (int)((n_dwords >> 16) & 0xFFFFu);     // tensor_dim0[31:16], tensor_dim1=0
    g1[3] = (int)((n_dwords & 0xFFFFu) << 16);     // tile_dim0 = n_dwords
    g1[4] = 0;                                     // tile_dim1/2 = 0 (1D tile)
    g1[5] = (int)n_dwords;                         // tensor_dim0_stride[31:0]
    g1[6] = 0;
    g1[7] = 0;
    i32x4 gz = {0, 0, 0, 0};
#if __has_include(<hip/amd_detail/amd_gfx1250_TDM.h>)
    i32x8 gz8 = {0, 0, 0, 0, 0, 0, 0, 0};
    __builtin_amdgcn_tensor_load_to_lds(g0, g1, gz, gz, gz8, 0);
#else
    __builtin_amdgcn_tensor_load_to_lds(g0, g1, gz, gz, 0);
#endif
  }
#else
  const int* s = (const int*)gsrc;
  int* d = (int*)lds_dst;
  for (unsigned k = tid; k < n_dwords; k += nthr) d[k] = s[k];
#endif
}

__device__ __forceinline__ void stage_wait(unsigned tid) {
#if HAVE_TDM
  if ((tid >> 5) == 0) {
#if __has_builtin(__builtin_amdgcn_s_wait_tensorcnt)
    __builtin_amdgcn_s_wait_tensorcnt(0);
#else
    asm volatile("s_wait_tensorcnt 0x0" ::: "memory");
#endif
  }
#endif
  __syncthreads();
}

// Fast fp32 reciprocal: v_rcp_f32 + 1 Newton-Raphson step (<=1 ulp).
__device__ __forceinline__ float fast_rcp(float x) {
#if defined(__AMDGCN__)
  float r = __builtin_amdgcn_rcpf(x);
#else
  float r = 1.0f / x;
#endif
  return r * (2.0f - x * r);
}

// ---------------------------------------------------------------------------
// Kernel A: per-proposal IoU row sweep, max/argmax over G, class labels.
// grid (PP/256, NN) x 64 threads; one thread = FOUR consecutive proposals so
// every IoU row write is one global_store_b128 (nontemporal: 134 MB streamed
// once, never re-read; keep it out of L2 for the later passes).
// ---------------------------------------------------------------------------
__global__ __launch_bounds__(64) void kA(const float4* __restrict__ gt4,
                                         const float4* __restrict__ prop4,
                                         const int* __restrict__ gtc,
                                         float* __restrict__ out) {
  __shared__ float4 s_gt[GG];  // 8 KB: all GT boxes for this image
  const int n = blockIdx.y;
  const int p0 = blockIdx.x * 256 + (int)threadIdx.x * 4;

  stage_to_lds(s_gt, gt4 + (size_t)n * GG, GG * 4, threadIdx.x, 64);
  stage_wait(threadIdx.x);

  float4 pb[4];
  float a2[4], best[4];
  int bidx[4];
#pragma unroll
  for (int k = 0; k < 4; ++k) {
    pb[k] = prop4[(size_t)n * PP + p0 + k];
    a2[k] = (pb[k].z - pb[k].x) * (pb[k].w - pb[k].y);
    best[k] = -1.0f;
    bidx[k] = 0;
  }

  float* __restrict__ iou_base = out + OFF_IOU + (size_t)n * GG * PP + p0;

  for (int g = 0; g < GG; ++g) {
    const float4 gb = s_gt[g];                     // LDS broadcast (all lanes same addr)
    const float area1 = (gb.z - gb.x) * (gb.w - gb.y);
    f32x4 v;
#pragma unroll
    for (int k = 0; k < 4; ++k) {
      const float lx = fmaxf(gb.x, pb[k].x);
      const float ly = fmaxf(gb.y, pb[k].y);
      const float rx = fminf(gb.z, pb[k].z);
      const float ry = fminf(gb.w, pb[k].w);
      const float w = fmaxf(rx - lx, 0.0f);
      const float h = fmaxf(ry - ly, 0.0f);
      const float inter = w * h;
      const float uni = area1 + a2[k] - inter;
      const float iou = (inter > 0.0f) ? inter * fast_rcp(uni) : 0.0f;
      v[k] = iou;
      if (iou > best[k]) { best[k] = iou; bidx[k] = g; }  // strict > keeps FIRST max
    }
    __builtin_nontemporal_store(v, (f32x4*)(iou_base + (size_t)g * PP));
  }

  f32x4 midxv, clsv;
#pragma unroll
  for (int k = 0; k < 4; ++k) {
    int cls;
    if (best[k] < 0.4f)      cls = 0;    // below BG_THRESH -> background
    else if (best[k] < 0.5f) cls = -1;   // between -> ignore
    else                     cls = gtc[n * GG + bidx[k]];
    midxv[k] = (float)bidx[k];
    clsv[k]  = (float)cls;
  }
  *(f32x4*)(out + OFF_MIDX + (size_t)n * PP + p0) = midxv;
  *(f32x4*)(out + OFF_CLS  + (size_t)n * PP + p0) = clsv;
}

// ---------------------------------------------------------------------------
// Kernel B1: stable count-ranks among fg and bg (reproduces stable argsort
// of where(mask, rand, 2.0)); scatter chosen samples into their final slot.
// grid (PP/256, NN) x 256 threads; one thread = one proposal; the O(P) scan
// reads LDS in float4 broadcasts (ds_load_b128).
// ---------------------------------------------------------------------------
__global__ __launch_bounds__(256) void kB1(const float* __restrict__ rnd,
                                           const float* __restrict__ cls_f,  // out[OFF_CLS]
                                           int* __restrict__ ws_inds,
                                           int* __restrict__ ws_unsel,
                                           int* __restrict__ ws_counts) {
  __shared__ float s_rand[PP];  // 16 KB
  __shared__ float s_cls[PP];   // 16 KB (cls as float: -1 / 0 / >0)
  const int n = blockIdx.y;
  const int i = blockIdx.x * 256 + (int)threadIdx.x;

  stage_to_lds(s_rand, rnd   + (size_t)n * PP, PP, threadIdx.x, 256);
  stage_to_lds(s_cls,  cls_f + (size_t)n * PP, PP, threadIdx.x, 256);
  stage_wait(threadIdx.x);

  const float ri = s_rand[i];
  const float ci = s_cls[i];
  const bool i_fg = (ci > 0.5f);
  const bool i_bg = (ci == 0.0f);

  int cnt_fg = 0, cnt_bg = 0, rfg = 0, rbg = 0;
  const f32x4* s_rand4 = (const f32x4*)s_rand;
  const f32x4* s_cls4  = (const f32x4*)s_cls;

  auto scan1 = [&](float rj, float cj, int j) {
    const int isfg = (cj > 0.5f) ? 1 : 0;
    const int isbg = (cj == 0.0f) ? 1 : 0;
    // Stable tie-break exactly like jnp.argsort: (r_j, j) < (r_i, i).
    const bool before = (rj < ri) || ((rj == ri) && (j < i));
    cnt_fg += isfg;
    cnt_bg += isbg;
    if (before) { rfg += isfg; rbg += isbg; }
  };

  for (int jv = 0; jv < PP / 4; ++jv) {
    const f32x4 r4 = s_rand4[jv];
    const f32x4 c4 = s_cls4[jv];
    const int j = jv * 4;
    scan1(r4[0], c4[0], j + 0);
    scan1(r4[1], c4[1], j + 1);
    scan1(r4[2], c4[2], j + 2);
    scan1(r4[3], c4[3], j + 3);
  }

  const int n_fg = min(cnt_fg, FG_CAP);
  const int n_bg = min(cnt_bg, BB - n_fg);

  bool chosen = false;
  int slot = 0;
  if (i_fg && rfg < n_fg)      { chosen = true; slot = rfg; }
  else if (i_bg && rbg < n_bg) { chosen = true; slot = n_fg + rbg; }

  if (chosen) ws_inds[n * BB + slot] = i;
  ws_unsel[(size_t)n * PP + i] = chosen ? 0 : 1;
  if (blockIdx.x == 0 && threadIdx.x == 0) {
    ws_counts[2 * n]     = n_fg;
    ws_counts[2 * n + 1] = n_bg;
  }
}

// ---------------------------------------------------------------------------
// Kernel B2: exclusive prefix scan of "unselected" flags -> fill padding
// slots [n_fg+n_bg, 512) with the lowest-index unselected proposals.
// grid NN x 256 threads; thread t owns the 16 contiguous flags [16t, 16t+16).
// ---------------------------------------------------------------------------
__global__ __launch_bounds__(256) void kB2(const int* __restrict__ ws_unsel,
                                           const int* __restrict__ ws_counts,
                                           int* __restrict__ ws_inds) {
  __shared__ int s_part[256];
  const int n = blockIdx.x;
  const int t = (int)threadIdx.x;
  const int base = t * 16;

  int loc[16];
  int sum = 0;
#pragma unroll
  for (int k = 0; k < 16; ++k) {
    loc[k] = ws_unsel[(size_t)n * PP + base + k];
    sum += loc[k];
  }
  s_part[t] = sum;
  __syncthreads();
  if (t == 0) {
    int acc = 0;
    for (int q = 0; q < 256; ++q) { const int v = s_part[q]; s_part[q] = acc; acc += v; }
  }
  __syncthreads();

  const int n_fg = ws_counts[2 * n];
  const int n_bg = ws_counts[2 * n + 1];
  const int basecnt = n_fg + n_bg;
  const int pad = BB - basecnt;

  int pre = s_part[t];
#pragma unroll
  for (int k = 0; k < 16; ++k) {
    if (loc[k]) {
      if (pre < pad) ws_inds[n * BB + basecnt + pre] = base + k;
      ++pre;
    }
  }
}

// ---------------------------------------------------------------------------
// Kernel C: gathers for the 512 sampled slots.
// grid NN x 512 threads; one thread = one sampled slot.
// ---------------------------------------------------------------------------
__global__ __launch_bounds__(512) void kC(const float4* __restrict__ gt4,
                                          const float4* __restrict__ prop4,
                                          const int* __restrict__ ws_inds,
                                          const int* __restrict__ ws_counts,
                                          float* __restrict__ out) {
  const int n = blockIdx.x;
  const int s = (int)threadIdx.x;
  const int ind  = ws_inds[n * BB + s];
  const int nsel = ws_counts[2 * n] + ws_counts[2 * n + 1];

  const float cls_f  = out[OFF_CLS  + (size_t)n * PP + ind];   // exact small ints as float
  const float midx_f = out[OFF_MIDX + (size_t)n * PP + ind];
  const int midx = (int)midx_f;

  out[OFF_SINDS + (size_t)n * BB + s] = (float)ind;
  out[OFF_SCLS  + (size_t)n * BB + s] = cls_f;

  const float4 gb  = gt4[(size_t)n * GG + midx];
  const float4 pbx = prop4[(size_t)n * PP + ind];
  ((float4*)(out + OFF_SGT))[(size_t)n * BB + s]   = gb;
  ((float4*)(out + OFF_SPROP))[(size_t)n * BB + s] = pbx;

  out[OFF_VALID + (size_t)n * BB + s] = (s < nsel) ? 1.0f : 0.0f;
}

// ---------------------------------------------------------------------------
extern "C" void kernel_launch(void* const* d_in, const int* in_sizes, int n_in,
                              void* d_out, int out_size, void* d_ws, size_t ws_size,
                              hipStream_t stream) {
  (void)in_sizes; (void)n_in; (void)out_size; (void)ws_size;
  const float4* gt4   = (const float4*)d_in[0];   // [N,G,4] f32
  const float4* prop4 = (const float4*)d_in[1];   // [N,P,4] f32
  const int*    gtc   = (const int*)d_in[2];      // [N,G]   i32
  const float*  rnd   = (const float*)d_in[3];    // [N,P]   f32
  float* out = (float*)d_out;

  // Workspace: unselected flags [N,P], slot->index [N,512], per-image counts.
  int* ws_unsel  = (int*)d_ws;                    // N*P ints   (256 KB)
  int* ws_inds   = ws_unsel + (size_t)NN * PP;    // N*512 ints ( 32 KB)
  int* ws_counts = ws_inds + (size_t)NN * BB;     // N*2 ints

  kA<<<dim3(PP / 256, NN), 64, 0, stream>>>(gt4, prop4, gtc, out);
  kB1<<<dim3(PP / 256, NN), 256, 0, stream>>>(rnd, out + OFF_CLS,
                                              ws_inds, ws_unsel, ws_counts);
  kB2<<<NN, 256, 0, stream>>>(ws_unsel, ws_counts, ws_inds);
  kC<<<NN, 512, 0, stream>>>(gt4, prop4, ws_inds, ws_counts, out);
}